// LocalGNNEncoder_39470749450709
// MI455X (gfx1250) — compile-verified
//
#include <hip/hip_runtime.h>
#include <hip/hip_bf16.h>
#include <math.h>

typedef __attribute__((ext_vector_type(2))) float v2f;
typedef __attribute__((ext_vector_type(8))) float v8f;

#define HEADS  4
#define NNODE  4096
#define ALPHA  0.2f
#define NEGV   (-9.0e15f)
#define CH     256   // attention column chunk staged through LDS

__device__ __forceinline__ float leaky(float v) { return v >= 0.f ? v : ALPHA * v; }

__device__ __forceinline__ float wave_sum(float v) {
#pragma unroll
  for (int off = 16; off > 0; off >>= 1) v += __shfl_xor(v, off, 32);
  return v;
}
__device__ __forceinline__ float wave_max(float v) {
#pragma unroll
  for (int off = 16; off > 0; off >>= 1) v = fmaxf(v, __shfl_xor(v, off, 32));
  return v;
}

// ---------------------------------------------------------------------------
// h[head] = X @ W[head]   (M x K) @ (K x Nd), fp32 WMMA 16x16x4.
// grid: (M/16, Nd/64, HEADS), block: 128 (4 waves, one 16x16 N-tile per wave)
// ---------------------------------------------------------------------------
__global__ void gat_gemm_xw(const float* __restrict__ X, const float* __restrict__ W,
                            float* __restrict__ H, int M, int K, int Nd) {
  const int wave = threadIdx.x >> 5;
  const int lane = threadIdx.x & 31;
  const int lo = lane & 15, hi = lane >> 4;
  const int head = blockIdx.z;
  const int row0 = blockIdx.x * 16;
  const int col0 = (blockIdx.y * 4 + wave) * 16;
  if (col0 >= Nd) return;  // wave-uniform

  const float* w = W + (size_t)head * K * Nd;
  float* h = H + (size_t)head * M * Nd;

  const int arow = row0 + lo;
  const int bcol = col0 + lo;
  v8f c = {};
  for (int k = 0; k < K; k += 4) {
    const int ka = k + 2 * hi;
    v2f a, b;
    a.x = X[(size_t)arow * K + ka];          // A: (M=lane%16, K = ka)
    a.y = X[(size_t)arow * K + ka + 1];
    b.x = w[(size_t)ka * Nd + bcol];         // B: (K = ka,   N=lane%16)
    b.y = w[(size_t)(ka + 1) * Nd + bcol];
    c = __builtin_amdgcn_wmma_f32_16x16x4_f32(false, a, false, b, (short)0, c,
                                              false, false);
  }
#pragma unroll
  for (int r = 0; r < 8; ++r)                // C/D: VGPR r -> M = r + 8*hi
    h[(size_t)(row0 + r + 8 * hi) * Nd + bcol] = c[r];
}

// ---------------------------------------------------------------------------
// e_src[i] = h[i,:] . a[:d],  e_dst[i] = h[i,:] . a[d:]
// grid: (N/8, 1, HEADS), block 256 (8 waves, one row per wave)
// ---------------------------------------------------------------------------
__global__ void gat_edge_scores(const float* __restrict__ H, const float* __restrict__ A,
                                float* __restrict__ esrc, float* __restrict__ edst, int d) {
  const int wave = threadIdx.x >> 5, lane = threadIdx.x & 31;
  const int head = blockIdx.z;
  const int row = blockIdx.x * 8 + wave;
  const float* h = H + ((size_t)head * NNODE + row) * d;
  const float* a = A + (size_t)head * 2 * d;
  float s0 = 0.f, s1 = 0.f;
  for (int n = lane; n < d; n += 32) {
    float v = h[n];
    s0 += v * a[n];
    s1 += v * a[d + n];
  }
  s0 = wave_sum(s0);
  s1 = wave_sum(s1);
  if (lane == 0) {
    esrc[head * NNODE + row] = s0;
    edst[head * NNODE + row] = s1;
  }
}

// ---------------------------------------------------------------------------
// Fused masked-softmax-attention:  O = softmax(mask(leaky(es_i + ed_j))) @ H
// One block = (16-row block, head), 256 threads = 8 wave32s.
//   Phase 1: per-row max & softmax denom (2 rows/wave, e_dst staged in LDS).
//   Phase 2: per 256-column chunk, all threads cooperatively compute the
//            16xCH probability tile ONCE (coalesced adj reads, one exp per
//            element) into LDS Pt[col][row] (stride 17 -> conflict-free),
//            then each wave assembles WMMA A-fragments from LDS and runs
//            fp32 v_wmma_16x16x4 against B-fragments of H.
// grid: (N/16, HEADS), block 256.
// ---------------------------------------------------------------------------
template <int D>
__global__ void gat_att_out(const float* __restrict__ H, const int* __restrict__ adj,
                            const float* __restrict__ esrc, const float* __restrict__ edst,
                            float* __restrict__ O) {
  const int tid = threadIdx.x, wave = tid >> 5, lane = tid & 31;
  const int lo = lane & 15, hi = lane >> 4;
  const int head = blockIdx.y;
  const int row0 = blockIdx.x * 16;
  const float* h = H + (size_t)head * NNODE * D;
  const float* es = esrc + head * NNODE;
  const float* ed = edst + head * NNODE;

  __shared__ float eds[NNODE];        // 16 KB: e_dst for this head
  __shared__ float Pt[CH][17];        // 17 KB: probability tile, transposed+padded
  __shared__ float sm[16], sl[16], sinvl[16];

  // stage e_dst (coalesced, read 3x afterwards)
  for (int c = tid; c < NNODE; c += 256) eds[c] = ed[c];
  __syncthreads();

  {  // phase 1: rows 2*wave, 2*wave+1
    const int r0 = row0 + 2 * wave, r1 = r0 + 1;
    const float e0 = es[r0], e1 = es[r1];
    const int* ar0 = adj + (size_t)r0 * NNODE;
    const int* ar1 = adj + (size_t)r1 * NNODE;
    float m0 = -INFINITY, m1 = -INFINITY;
    for (int c = lane; c < NNODE; c += 32) {
      const float dd = eds[c];
      m0 = fmaxf(m0, ar0[c] ? leaky(e0 + dd) : NEGV);
      m1 = fmaxf(m1, ar1[c] ? leaky(e1 + dd) : NEGV);
    }
    m0 = wave_max(m0);
    m1 = wave_max(m1);
    float l0 = 0.f, l1 = 0.f;
    for (int c = lane; c < NNODE; c += 32) {
      const float dd = eds[c];
      l0 += __expf((ar0[c] ? leaky(e0 + dd) : NEGV) - m0);
      l1 += __expf((ar1[c] ? leaky(e1 + dd) : NEGV) - m1);
    }
    l0 = wave_sum(l0);
    l1 = wave_sum(l1);
    if (lane == 0) {
      sm[2 * wave] = m0;
      sm[2 * wave + 1] = m1;
      sl[2 * wave] = l0;
      sl[2 * wave + 1] = l1;
    }
  }
  __syncthreads();
  if (tid < 16) sinvl[tid] = 1.0f / sl[tid];

  // phase 2
  constexpr int TPW = D / 128;  // N-tiles per wave (8 waves cover D/16 tiles)
  v8f acc[TPW];
#pragma unroll
  for (int t = 0; t < TPW; ++t) acc[t] = (v8f){};

  for (int c0 = 0; c0 < NNODE; c0 += CH) {
    __syncthreads();  // previous chunk's Pt reads complete
    // cooperative P-tile staging: thread tid owns column c0+tid
    const float dd = eds[c0 + tid];
#pragma unroll 4
    for (int r = 0; r < 16; ++r) {
      const int* arow = adj + (size_t)(row0 + r) * NNODE + c0;
      __builtin_prefetch(arow + CH, 0, 1);  // next chunk of this adj row
      const float s = arow[tid] ? leaky(es[row0 + r] + dd) : NEGV;
      Pt[tid][r] = __expf(s - sm[r]) * sinvl[r];
    }
    __syncthreads();  // Pt visible

    for (int kk = 0; kk < CH; kk += 4) {
      const int k0 = kk + 2 * hi;
      v2f a;
      a.x = Pt[k0][lo];       // A: (M=lo, K=k0)   conflict-free (stride 17)
      a.y = Pt[k0 + 1][lo];
#pragma unroll
      for (int t = 0; t < TPW; ++t) {
        const int col = (wave * TPW + t) * 16 + lo;
        v2f b;
        b.x = h[(size_t)(c0 + k0) * D + col];      // B: (K, N=col)
        b.y = h[(size_t)(c0 + k0 + 1) * D + col];
        acc[t] = __builtin_amdgcn_wmma_f32_16x16x4_f32(false, a, false, b,
                                                       (short)0, acc[t],
                                                       false, false);
      }
    }
  }

  float* o = O + (size_t)head * NNODE * D;
#pragma unroll
  for (int t = 0; t < TPW; ++t) {
    const int col = (wave * TPW + t) * 16 + lo;
#pragma unroll
    for (int r = 0; r < 8; ++r)
      o[(size_t)(row0 + r + 8 * hi) * D + col] = acc[t][r];
  }
}

// ---------------------------------------------------------------------------
// Per-row post: leaky -> L2 normalize -> +bias -> mean over heads -> (elu)
// grid: N, block: D threads.
// ---------------------------------------------------------------------------
template <int D, bool ELU>
__global__ void gat_post(const float* __restrict__ O, const float* __restrict__ Bq,
                         float* __restrict__ Xout) {
  const int row = blockIdx.x, t = threadIdx.x;
  const int wave = t >> 5, lane = t & 31;
  __shared__ float red[D / 32];
  float acc = 0.f;
#pragma unroll
  for (int hd = 0; hd < HEADS; ++hd) {
    float v = O[((size_t)hd * NNODE + row) * D + t];
    v = leaky(v);
    const float ss = wave_sum(v * v);
    if (lane == 0) red[wave] = ss;
    __syncthreads();
    float tot = 0.f;
#pragma unroll
    for (int i = 0; i < D / 32; ++i) tot += red[i];
    __syncthreads();
    const float nrm = sqrtf(tot);
    acc += v / fmaxf(nrm, 1e-12f) + Bq[hd * D + t];
  }
  acc *= (1.0f / HEADS);
  if (ELU) acc = acc > 0.f ? acc : expm1f(acc);
  Xout[(size_t)row * D + t] = acc;
}

// ---------------------------------------------------------------------------
extern "C" void kernel_launch(void* const* d_in, const int* in_sizes, int n_in,
                              void* d_out, int out_size, void* d_ws, size_t ws_size,
                              hipStream_t stream) {
  (void)in_sizes; (void)n_in; (void)out_size; (void)ws_size;
  const float* x   = (const float*)d_in[0];   // [4096,512]
  const int*   adj = (const int*)  d_in[1];   // [4096,4096]
  const float* W1  = (const float*)d_in[2];   // [4,512,256]
  const float* a1  = (const float*)d_in[3];   // [4,512]
  const float* b1  = (const float*)d_in[4];   // [4,256]
  const float* W2  = (const float*)d_in[5];   // [4,256,128]
  const float* a2  = (const float*)d_in[6];   // [4,256]
  const float* b2  = (const float*)d_in[7];   // [4,128]
  float* out = (float*)d_out;                 // [4096,128]

  float* ws = (float*)d_ws;
  const size_t HB = (size_t)HEADS * NNODE * 256;  // 4,194,304 floats
  float* h1  = ws;                 // [4,4096,256]  (layer2 h2 aliases here)
  float* o1  = ws + HB;            // [4,4096,256]  (layer2 o2 aliases here)
  float* x1  = ws + 2 * HB;        // [4096,256]
  float* es1 = x1 + (size_t)NNODE * 256;
  float* ed1 = es1 + HEADS * NNODE;
  float* es2 = ed1 + HEADS * NNODE;
  float* ed2 = es2 + HEADS * NNODE;
  float* h2 = h1;  // reuse (h1 dead after x1 computed)
  float* o2 = o1;  // reuse (o1 dead after x1 computed)

  // ---- layer 1 (din=512, d=256) ----
  gat_gemm_xw<<<dim3(NNODE / 16, 256 / 64, HEADS), 128, 0, stream>>>(
      x, W1, h1, NNODE, 512, 256);
  gat_edge_scores<<<dim3(NNODE / 8, 1, HEADS), 256, 0, stream>>>(
      h1, a1, es1, ed1, 256);
  gat_att_out<256><<<dim3(NNODE / 16, HEADS), 256, 0, stream>>>(
      h1, adj, es1, ed1, o1);
  gat_post<256, true><<<NNODE, 256, 0, stream>>>(o1, b1, x1);

  // ---- layer 2 (din=256, d=128) ----
  gat_gemm_xw<<<dim3(NNODE / 16, 128 / 64, HEADS), 128, 0, stream>>>(
      x1, W2, h2, NNODE, 256, 128);
  gat_edge_scores<<<dim3(NNODE / 8, 1, HEADS), 256, 0, stream>>>(
      h2, a2, es2, ed2, 128);
  gat_att_out<128><<<dim3(NNODE / 16, HEADS), 256, 0, stream>>>(
      h2, adj, es2, ed2, o2);
  gat_post<128, false><<<NNODE, 128, 0, stream>>>(o2, b2, out);
}